// CapsuleLayer_1219770712153
// MI455X (gfx1250) — compile-verified
//
#include <hip/hip_runtime.h>

typedef __attribute__((ext_vector_type(2))) float v2f;
typedef __attribute__((ext_vector_type(8))) float v8f;

#define BATCH  128
#define JCAP   2048
#define IDIM   8
#define KCAP   32
#define DDIM   16
#define BT     16           // batch tile (WMMA M)
#define NCHUNK 16           // j-chunks
#define JC     (JCAP / NCHUNK)

// One workgroup: 16 batches x one j-chunk. 8 waves; wave w computes u_hat for
// k = 4w..4w+3 via V_WMMA_F32_16X16X4_F32 (two K=4 steps over I=8), parks the
// [K=32][B=16][D=16] tile in LDS, then 256 threads handle 512 (b,k) pairs:
// logit = V_acc . u_hat, softmax over K, accumulate partial s in registers.
__global__ __launch_bounds__(256) void caps_uhat_route_partial(
    const float* __restrict__ x, const float* __restrict__ Wt,
    const float* __restrict__ Vacc, float* __restrict__ spart, int first)
{
  __shared__ float U[KCAP][BT][DDIM];     // 32 KB u_hat tile for current j
  __shared__ float L[BT][KCAP + 1];       // padded logits (conflict-free)

  const int t     = threadIdx.x;
  const int wave  = t >> 5;
  const int lane  = t & 31;
  const int m     = lane & 15;            // WMMA row / col index
  const int khalf = lane >> 4;            // K half select per A/B layout

  const int b0 = blockIdx.y * BT;
  const int j0 = blockIdx.x * JC;

  // routing-stage mapping: thread owns (bl, k0) and (bl, k0+16)
  const int bl = t & 15;
  const int k0 = t >> 4;
  const int k1 = k0 + 16;

  float s0[DDIM], s1[DDIM], va0[DDIM], va1[DDIM];
#pragma unroll
  for (int d = 0; d < DDIM; ++d) { s0[d] = 0.f; s1[d] = 0.f; va0[d] = 0.f; va1[d] = 0.f; }

  if (!first) {
    const float4* p0 = (const float4*)&Vacc[((size_t)(b0 + bl) * KCAP + k0) * DDIM];
    const float4* p1 = (const float4*)&Vacc[((size_t)(b0 + bl) * KCAP + k1) * DDIM];
#pragma unroll
    for (int q = 0; q < 4; ++q) {
      float4 a = p0[q]; va0[4*q] = a.x; va0[4*q+1] = a.y; va0[4*q+2] = a.z; va0[4*q+3] = a.w;
      float4 b = p1[q]; va1[4*q] = b.x; va1[4*q+1] = b.y; va1[4*q+2] = b.z; va1[4*q+3] = b.w;
    }
  }

  for (int jj = 0; jj < JC; ++jj) {
    const int j = j0 + jj;

    // A fragment: X_j[16 x 8]; lane=M, {vgpr,lane-half}=K (ISA 16x4 f32 layout)
    const float* xr = &x[((size_t)(b0 + m) * JCAP + j) * IDIM + 2 * khalf];
    v2f a0, a1;
    a0.x = xr[0]; a0.y = xr[1];       // i = 2*khalf, 2*khalf+1
    a1.x = xr[4]; a1.y = xr[5];       // i = 2*khalf+4, 2*khalf+5

#pragma unroll
    for (int kk = 0; kk < 4; ++kk) {
      const int k = wave * 4 + kk;
      // B fragment: W_{j,k}[8 x 16]; lane=N, {vgpr,lane-half}=K
      const float* wr = &Wt[(((size_t)j * KCAP + k) * IDIM) * DDIM + m];
      v2f bf0, bf1;
      bf0.x = wr[(2 * khalf + 0) * DDIM];
      bf0.y = wr[(2 * khalf + 1) * DDIM];
      bf1.x = wr[(2 * khalf + 4) * DDIM];
      bf1.y = wr[(2 * khalf + 5) * DDIM];

      v8f acc = {};
      acc = __builtin_amdgcn_wmma_f32_16x16x4_f32(false, a0, false, bf0, (short)0, acc, false, false);
      acc = __builtin_amdgcn_wmma_f32_16x16x4_f32(false, a1, false, bf1, (short)0, acc, false, false);

      // D layout: vgpr r, lane-half h -> row r + 8h, col m
#pragma unroll
      for (int r = 0; r < 8; ++r) U[k][r + 8 * khalf][m] = acc[r];
    }
    __syncthreads();

    float c0, c1;
    float l0 = 0.f, l1 = 0.f;
    if (first) {
      c0 = c1 = (1.0f / 32.0f);       // softmax of zero logits
    } else {
      const float4* u0 = (const float4*)&U[k0][bl][0];
      const float4* u1 = (const float4*)&U[k1][bl][0];
#pragma unroll
      for (int q = 0; q < 4; ++q) {
        float4 a = u0[q];
        l0 += a.x * va0[4*q] + a.y * va0[4*q+1] + a.z * va0[4*q+2] + a.w * va0[4*q+3];
        float4 b = u1[q];
        l1 += b.x * va1[4*q] + b.y * va1[4*q+1] + b.z * va1[4*q+2] + b.w * va1[4*q+3];
      }
      L[bl][k0] = l0; L[bl][k1] = l1;
      __syncthreads();
      float mx = -1e30f;
#pragma unroll
      for (int kq = 0; kq < KCAP; ++kq) mx = fmaxf(mx, L[bl][kq]);
      float se = 0.f;
#pragma unroll
      for (int kq = 0; kq < KCAP; ++kq) se += __expf(L[bl][kq] - mx);
      const float inv = 1.0f / se;
      c0 = __expf(l0 - mx) * inv;
      c1 = __expf(l1 - mx) * inv;
    }

    {
      const float4* u0 = (const float4*)&U[k0][bl][0];
      const float4* u1 = (const float4*)&U[k1][bl][0];
#pragma unroll
      for (int q = 0; q < 4; ++q) {
        float4 a = u0[q];
        s0[4*q] += c0 * a.x; s0[4*q+1] += c0 * a.y; s0[4*q+2] += c0 * a.z; s0[4*q+3] += c0 * a.w;
        float4 b = u1[q];
        s1[4*q] += c1 * b.x; s1[4*q+1] += c1 * b.y; s1[4*q+2] += c1 * b.z; s1[4*q+3] += c1 * b.w;
      }
    }
    __syncthreads();
  }

  float4* o0 = (float4*)&spart[(((size_t)blockIdx.x * BATCH + (b0 + bl)) * KCAP + k0) * DDIM];
  float4* o1 = (float4*)&spart[(((size_t)blockIdx.x * BATCH + (b0 + bl)) * KCAP + k1) * DDIM];
#pragma unroll
  for (int q = 0; q < 4; ++q) {
    o0[q] = make_float4(s0[4*q], s0[4*q+1], s0[4*q+2], s0[4*q+3]);
    o1[q] = make_float4(s1[4*q], s1[4*q+1], s1[4*q+2], s1[4*q+3]);
  }
}

// Reduce j-chunks, squash, then: mode 0 -> Vacc = v, mode 1 -> Vacc += v,
// mode 2 -> out = v.
__global__ __launch_bounds__(256) void caps_squash(
    const float* __restrict__ spart, float* __restrict__ Vacc,
    float* __restrict__ out, int mode)
{
  const int gid = blockIdx.x * 256 + threadIdx.x;   // 0..4095 -> (b,k)
  const int b = gid >> 5;
  const int k = gid & 31;

  float s[DDIM];
#pragma unroll
  for (int d = 0; d < DDIM; ++d) s[d] = 0.f;
  for (int c = 0; c < NCHUNK; ++c) {
    const float4* p = (const float4*)&spart[(((size_t)c * BATCH + b) * KCAP + k) * DDIM];
#pragma unroll
    for (int q = 0; q < 4; ++q) {
      float4 a = p[q];
      s[4*q] += a.x; s[4*q+1] += a.y; s[4*q+2] += a.z; s[4*q+3] += a.w;
    }
  }

  float s2 = 0.f;
#pragma unroll
  for (int d = 0; d < DDIM; ++d) s2 += s[d] * s[d];
  const float scale = (s2 / (1.0f + s2)) * rsqrtf(s2 + 1e-7f);

  float* dst = (mode == 2) ? &out[((size_t)b * KCAP + k) * DDIM]
                           : &Vacc[((size_t)b * KCAP + k) * DDIM];
  if (mode == 1) {
#pragma unroll
    for (int d = 0; d < DDIM; ++d) dst[d] += scale * s[d];
  } else {
#pragma unroll
    for (int d = 0; d < DDIM; ++d) dst[d] = scale * s[d];
  }
}

extern "C" void kernel_launch(void* const* d_in, const int* in_sizes, int n_in,
                              void* d_out, int out_size, void* d_ws, size_t ws_size,
                              hipStream_t stream) {
  (void)in_sizes; (void)n_in; (void)out_size; (void)ws_size;
  const float* x  = (const float*)d_in[0];   // [128, 2048, 8] f32
  const float* Wt = (const float*)d_in[1];   // [2048, 32, 8, 16] f32
  float* out = (float*)d_out;                // [128, 32, 16] f32

  // workspace: spart [NCHUNK][B][K][D] (4 MB) + Vacc [B][K][D] (256 KB)
  float* spart = (float*)d_ws;
  float* Vacc  = spart + (size_t)NCHUNK * BATCH * KCAP * DDIM;

  dim3 gA(NCHUNK, BATCH / BT, 1), bA(256, 1, 1);
  dim3 gB((BATCH * KCAP) / 256, 1, 1), bB(256, 1, 1);

  // iteration 0: uniform coupling (softmax of zeros), Vacc = v
  caps_uhat_route_partial<<<gA, bA, 0, stream>>>(x, Wt, Vacc, spart, 1);
  caps_squash<<<gB, bB, 0, stream>>>(spart, Vacc, out, 0);
  // iteration 1: logits = Vacc . u_hat, Vacc += v
  caps_uhat_route_partial<<<gA, bA, 0, stream>>>(x, Wt, Vacc, spart, 0);
  caps_squash<<<gB, bB, 0, stream>>>(spart, Vacc, out, 1);
  // iteration 2: final v -> out
  caps_uhat_route_partial<<<gA, bA, 0, stream>>>(x, Wt, Vacc, spart, 0);
  caps_squash<<<gB, bB, 0, stream>>>(spart, Vacc, out, 2);
}